// DetectionLoss_89550068121905
// MI455X (gfx1250) — compile-verified
//
#include <hip/hip_runtime.h>
#include <hip/hip_bf16.h>

// ---------------- problem constants ----------------
#define B_    16
#define C_    85
#define H_    128
#define W_    128
#define HW_   (H_ * W_)          // 16384
#define NC_   (C_ - 5)           // 80
#define NT_   64                 // targets per image
// streaming region: channels 4..84 per image, contiguous 81*HW floats
#define REG_FLOATS (81 * HW_)    // 1,327,104
#define CHUNK_FLOATS 1024        // 256 threads * 4 floats
#define CHUNK_BYTES  4096
#define CPI  (REG_FLOATS / CHUNK_FLOATS)   // 1296 chunks per image
#define NCHUNK (B_ * CPI)                  // 20736
#define K1_GRID  2048
#define K1_BLOCK 256
#define NSTAGE   4               // LDS stages; 3 kept in flight
#define P_  2048                 // partial count (== K1_GRID, mult of 64)

typedef __attribute__((ext_vector_type(2))) float v2f;
typedef __attribute__((ext_vector_type(8))) float v8f;

// fast, numerically-stable softplus: max(x,0) + log(1 + exp(-|x|))
__device__ __forceinline__ float softplus_f(float x) {
    float ax = __builtin_fabsf(x);
    return fmaxf(x, 0.0f) + __logf(1.0f + __expf(-ax));
}

// ============================================================
// Kernel 1: stream channels 4..84 of every image through LDS via
// CDNA5 async global->LDS loads. 4 stages, 3 async ops in flight
// per wave (3 x 512B x 16384 waves = 24 MB >= 23.3 TB/s x ~1us).
// ws[bid] = obj partial, ws[P_ + bid] = cls partial.
// ============================================================
__global__ __launch_bounds__(K1_BLOCK) void k1_softplus_stream(
    const float* __restrict__ preds, float* __restrict__ ws) {

    __shared__ __align__(16) float buf[NSTAGE * CHUNK_FLOATS];   // 16 KB

    const unsigned tid = threadIdx.x;
    const unsigned bid = blockIdx.x;
    const unsigned stride = gridDim.x;

    // LDS byte offset of buf (low 32 bits of the generic pointer)
    const unsigned ldsBase = (unsigned)(unsigned long long)(size_t)(&buf[0]);

    float s_obj = 0.0f, s_cls = 0.0f;

    auto issue = [&](unsigned i /*local chunk #*/) {
        unsigned c   = bid + i * stride;        // global chunk id
        unsigned img = c / (unsigned)CPI;
        unsigned r   = c % (unsigned)CPI;
        const float* src = preds
            + (size_t)img * (C_ * HW_) + (size_t)4 * HW_
            + (size_t)r * CHUNK_FLOATS + (size_t)tid * 4;
        unsigned lds = ldsBase + (i & (NSTAGE - 1)) * CHUNK_BYTES + tid * 16u;
        unsigned long long gaddr = (unsigned long long)src;
        asm volatile("global_load_async_to_lds_b128 %0, %1, off"
                     :: "v"(lds), "v"(gaddr) : "memory");
    };

    auto consume = [&](unsigned i /*local chunk #*/) {
        unsigned c = bid + i * stride;
        const float4 d = *reinterpret_cast<const float4*>(
            &buf[(i & (NSTAGE - 1)) * CHUNK_FLOATS + tid * 4]);
        float s4 = softplus_f(d.x) + softplus_f(d.y)
                 + softplus_f(d.z) + softplus_f(d.w);
        if ((c % (unsigned)CPI) < 16u) s_obj += s4;   // channel-4 plane
        else                           s_cls += s4;   // channels 5..84
    };

    // number of chunks handled by this block (block-uniform; bid < NCHUNK always)
    const unsigned nch = (NCHUNK - 1u - bid) / stride + 1u;

    // prologue: put up to 3 async loads in flight
    unsigned issued = 0;
    const unsigned pre = (nch < NSTAGE - 1u) ? nch : (NSTAGE - 1u);
    for (; issued < pre; ++issued) issue(issued);

    // steady state: 3 in flight; wait <=2 => oldest complete; consume; refill.
    // Refill goes to stage (i+3)&3, consumed at iteration i-1 => no LDS
    // read/write overlap (async LDS writes are unordered vs DS reads).
    unsigned i = 0;
    for (; issued < nch; ++issued, ++i) {
        asm volatile("s_wait_asynccnt 0x2" ::: "memory");
        consume(i);
        issue(issued);
    }
    // tail: drain everything, consume the last <=3 chunks
    asm volatile("s_wait_asynccnt 0x0" ::: "memory");
    for (; i < nch; ++i) consume(i);

    // wave reduce (wave32), then cross-wave via LDS (fixed order -> deterministic)
    #pragma unroll
    for (int off = 16; off > 0; off >>= 1) {
        s_obj += __shfl_xor(s_obj, off, 32);
        s_cls += __shfl_xor(s_cls, off, 32);
    }
    __syncthreads();                       // pipeline fully drained; reuse buf
    if ((tid & 31u) == 0u) {
        buf[tid >> 5] = s_obj;
        buf[8 + (tid >> 5)] = s_cls;
    }
    __syncthreads();
    if (tid == 0) {
        float o = 0.0f, cl = 0.0f;
        #pragma unroll
        for (int j = 0; j < K1_BLOCK / 32; ++j) { o += buf[j]; cl += buf[8 + j]; }
        ws[bid]      = o;
        ws[P_ + bid] = cl;
    }
}

// ============================================================
// Kernel 2: per-target work (B*N = 1024 threads, 1 block).
// IoU box loss (all targets) + dedup'd -x corrections for the
// scatter-set semantics of obj/cls targets.
// ws[2P_+0]=box_sum, ws[2P_+1]=corr_obj, ws[2P_+2]=corr_cls
// ============================================================
__global__ __launch_bounds__(1024) void k2_targets(
    const float* __restrict__ preds, const float* __restrict__ targets,
    float* __restrict__ ws) {

    __shared__ int   sIdx[B_ * NT_];
    __shared__ int   sCls[B_ * NT_];
    __shared__ float red[B_ * NT_];

    const int t = threadIdx.x;             // 0..1023
    const int b = t >> 6;
    const float* tg = targets + (size_t)t * 5;

    float clsf = tg[0], cx = tg[1], cy = tg[2], w = tg[3], h = tg[4];
    int gi  = (int)(cx * (float)W_);
    int gj  = (int)(cy * (float)H_);
    int idx = gj * W_ + gi;
    int cls = (int)clsf;

    sIdx[t] = idx; sCls[t] = cls;
    __syncthreads();

    bool firstObj = true, firstCls = true;
    for (int k = b * NT_; k < t; ++k) {    // earlier targets in same image
        if (sIdx[k] == idx) {
            firstObj = false;
            if (sCls[k] == cls) firstCls = false;
        }
    }

    const float* pbp = preds + (size_t)b * (C_ * HW_) + idx;
    float px = pbp[0];
    float py = pbp[HW_];
    float pw = pbp[2 * (size_t)HW_];
    float ph = pbp[3 * (size_t)HW_];
    float pobj = pbp[4 * (size_t)HW_];
    float pcls = pbp[(size_t)(5 + cls) * HW_];

    float x1 = px - pw * 0.5f, y1 = py - ph * 0.5f;
    float x2 = px + pw * 0.5f, y2 = py + ph * 0.5f;
    float gx1 = (cx - w * 0.5f) * (float)W_, gy1 = (cy - h * 0.5f) * (float)H_;
    float gx2 = (cx + w * 0.5f) * (float)W_, gy2 = (cy + h * 0.5f) * (float)H_;
    float ix1 = fmaxf(x1, gx1), iy1 = fmaxf(y1, gy1);
    float ix2 = fminf(x2, gx2), iy2 = fminf(y2, gy2);
    float inter = fmaxf(ix2 - ix1, 0.0f) * fmaxf(iy2 - iy1, 0.0f);
    float a1 = (x2 - x1) * (y2 - y1);
    float a2 = (gx2 - gx1) * (gy2 - gy1);
    float iou = inter / (a1 + a2 - inter + 1e-7f);

    float vbox = 1.0f - iou;
    float vco  = firstObj ? pobj : 0.0f;
    float vcc  = firstCls ? pcls : 0.0f;

    // three deterministic LDS tree reductions over 1024 threads
    float vals[3] = {vbox, vco, vcc};
    #pragma unroll
    for (int j = 0; j < 3; ++j) {
        __syncthreads();
        red[t] = vals[j];
        __syncthreads();
        for (int s = 512; s > 0; s >>= 1) {
            if (t < s) red[t] += red[t + s];
            __syncthreads();
        }
        if (t == 0) ws[2 * P_ + j] = red[0];
    }
}

// ============================================================
// Kernel 3: one wave32. Fold the 2*P_ block partials on the matrix
// pipe with V_WMMA_F32_16X16X4_F32 (D = A x ones + C => every column
// of D holds the grand total after one shfl), then combine losses.
// ============================================================
__global__ __launch_bounds__(32) void k3_finalize(
    const float* __restrict__ ws, float* __restrict__ out) {

    const int lane = threadIdx.x;

    v2f ones; ones[0] = 1.0f; ones[1] = 1.0f;   // B = all-ones 4x16
    v8f cObj = {}; v8f cCls = {};

    for (int i = 0; i < P_; i += 64) {
        v2f a; a[0] = ws[i + lane]; a[1] = ws[i + 32 + lane];
        cObj = __builtin_amdgcn_wmma_f32_16x16x4_f32(
            false, a, false, ones, (short)0, cObj, false, false);
    }
    for (int i = 0; i < P_; i += 64) {
        v2f a; a[0] = ws[P_ + i + lane]; a[1] = ws[P_ + i + 32 + lane];
        cCls = __builtin_amdgcn_wmma_f32_16x16x4_f32(
            false, a, false, ones, (short)0, cCls, false, false);
    }

    float so = cObj[0] + cObj[1] + cObj[2] + cObj[3]
             + cObj[4] + cObj[5] + cObj[6] + cObj[7];
    float sc = cCls[0] + cCls[1] + cCls[2] + cCls[3]
             + cCls[4] + cCls[5] + cCls[6] + cCls[7];
    so += __shfl_xor(so, 16, 32);   // column sum == grand total on every lane
    sc += __shfl_xor(sc, 16, 32);

    if (lane == 0) {
        float box = ws[2 * P_ + 0];
        float co  = ws[2 * P_ + 1];
        float cc  = ws[2 * P_ + 2];
        float obj_loss = (so - co) / (float)HW_;
        float cls_loss = (sc - cc) / ((float)HW_ * (float)NC_);
        out[0] = 0.05f * box + 1.0f * obj_loss + 0.5f * cls_loss;
    }
}

// ============================================================
extern "C" void kernel_launch(void* const* d_in, const int* in_sizes, int n_in,
                              void* d_out, int out_size, void* d_ws, size_t ws_size,
                              hipStream_t stream) {
    const float* preds   = (const float*)d_in[0];
    const float* targets = (const float*)d_in[1];
    float* out = (float*)d_out;
    float* ws  = (float*)d_ws;

    k1_softplus_stream<<<K1_GRID, K1_BLOCK, 0, stream>>>(preds, ws);
    k2_targets<<<1, B_ * NT_, 0, stream>>>(preds, targets, ws);
    k3_finalize<<<1, 32, 0, stream>>>(ws, out);
}